// STAttentionBlock_80126909874770
// MI455X (gfx1250) — compile-verified
//
#include <hip/hip_runtime.h>
#include <hip/hip_bf16.h>
#include <math.h>

// ---------------------------------------------------------------------------
// ST-Attention block for MI455X (gfx1250)
//   - fp32 WMMA (V_WMMA_F32_16X16X4_F32) for every GEMM
//   - Tensor Data Mover (TENSOR_LOAD_TO_LDS) for LDS panel staging,
//     double-buffered against the WMMA stream, synced with s_wait_tensorcnt
// N=32, C=OC=256, IC=64, S=3, T=64, V=25, J = T*V = 1600
// ---------------------------------------------------------------------------

typedef float v2f __attribute__((ext_vector_type(2)));
typedef float v8f __attribute__((ext_vector_type(8)));
typedef unsigned int v4u __attribute__((ext_vector_type(4)));
typedef int v4i_t __attribute__((ext_vector_type(4)));
typedef int v8i_t __attribute__((ext_vector_type(8)));

__device__ __forceinline__ v8f wmma_f32(v2f a, v2f b, v8f c) {
  // 8 args: (neg_a, A, neg_b, B, c_mod, C, reuse_a, reuse_b)
  return __builtin_amdgcn_wmma_f32_16x16x4_f32(false, a, false, b, (short)0, c,
                                               false, false);
}

#define BN_SCALE 0.9999950000374997f  // 1/sqrt(1+1e-5)

// ---- Tensor Data Mover availability (device pass only) --------------------
#if defined(__has_builtin)
#if __has_builtin(__builtin_amdgcn_tensor_load_to_lds) && \
    __has_builtin(__builtin_amdgcn_s_wait_tensorcnt)
#define HAVE_TDM 1
#endif
#endif
#ifndef HAVE_TDM
#define HAVE_TDM 0
#endif

#if HAVE_TDM
// 2D TDM load: width x height dwords, row stride in dwords, with LDS padding
// pad_int_code: dwords-per-pad-interval = 2<<code ; pad_amt_code: pad dwords-1
__device__ __forceinline__ void tdm_load_2d(const void* gptr, unsigned lds_off,
                                            unsigned width, unsigned height,
                                            unsigned stride,
                                            unsigned pad_int_code,
                                            unsigned pad_amt_code) {
  unsigned long long ga = (unsigned long long)(uintptr_t)gptr;
  v4u g0;
  g0.x = 1u;  // count=1, is_restore=0, gather_mode=0
  g0.y = lds_off;
  g0.z = (unsigned)(ga & 0xFFFFFFFFu);
  g0.w = (unsigned)((ga >> 32) & 0x1FFFFFFu) | (2u << 30);  // type=2 (image)
  v8i_t g1;
  // data_size=2 (4B), pad_enable, pad interval/amount; workgroup_mask=0
  g1[0] = (int)((2u << 16) | (1u << 20) | (pad_int_code << 22) |
                (pad_amt_code << 25));
  g1[1] = (int)((width & 0xFFFFu) << 16);   // tensor_dim0 lo16 @ bits 63:48
  g1[2] = (int)(((width >> 16) & 0xFFFFu) | ((height & 0xFFFFu) << 16));
  g1[3] = (int)(((height >> 16) & 0xFFFFu) | ((width & 0xFFFFu) << 16));
  g1[4] = (int)(height & 0xFFFFu);          // tile_dim1 ; tile_dim2=0
  g1[5] = (int)stride;                      // tensor_dim0_stride lo32
  g1[6] = 0;
  g1[7] = 0;
  v4i_t z4 = {0, 0, 0, 0};                    // groups 2/3 unused (2D tensor)
  v8i_t z8 = {0, 0, 0, 0, 0, 0, 0, 0};        // extra group (6-arg builtin)
  __builtin_amdgcn_tensor_load_to_lds(g0, g1, z4, z4, z8, 0);
}
#define LDS_OFF(p) ((unsigned)(uintptr_t)(const void*)(p))
#endif

// ---------------------------------------------------------------------------
// Generic conv1x1 GEMM:  OUT[b,m,j] = epi( sum_k W[m,k] * IN[b,k,j] )
// epi: +bias[m]; *gamma[m]*BN_SCALE + beta[m]; +res[b,m,j]; leaky-relu(0.1)
// Block = 128 thr (4 waves); 64(M) x 64(J) tile, BK=32.
// TDM double-buffers W(64x32->pitch36) and IN(32x64->pitch65) panels.
// ---------------------------------------------------------------------------
__global__ void __launch_bounds__(128) conv1x1_wmma(
    const float* __restrict__ W, const float* __restrict__ IN,
    float* __restrict__ OUT, const float* __restrict__ bias,
    const float* __restrict__ gamma, const float* __restrict__ beta,
    const float* __restrict__ res, int M, int K, int J, int act) {
  __shared__ float Ws[2][64][36];  // [m][k], pitch 36 (bank-safe)
  __shared__ float Bs[2][32][65];  // [k][j], pitch 65

  const int tid = threadIdx.x;
  const int lane = tid & 31;
  const int wave = tid >> 5;
  const int j0 = blockIdx.x * 64;
  const int m0 = blockIdx.y * 64;
  const int b = blockIdx.z;
  const float* INb = IN + (size_t)b * K * J;

  v8f acc[4] = {};

  const int col = lane & 15;
  const int koff = (lane >> 4) * 2;
  const int mrow = wave * 16 + col;  // A-fragment row (local)
  const int nkb = K >> 5;

#if HAVE_TDM
  if (wave == 0) {  // one TDM issue per workgroup panel
    tdm_load_2d(W + (size_t)m0 * K, LDS_OFF(&Ws[0][0][0]), 32u, 64u,
                (unsigned)K, 4u, 3u);
    tdm_load_2d(INb + j0, LDS_OFF(&Bs[0][0][0]), 64u, 32u, (unsigned)J, 5u,
                0u);
  }
  int buf = 0;
  for (int ib = 0; ib < nkb; ++ib) {
    if (wave == 0) __builtin_amdgcn_s_wait_tensorcnt(0);
    __syncthreads();
    if (wave == 0 && ib + 1 < nkb) {  // prefetch next panel via TDM
      int kb2 = (ib + 1) << 5;
      tdm_load_2d(W + (size_t)m0 * K + kb2,
                  LDS_OFF(&Ws[buf ^ 1][0][0]), 32u, 64u, (unsigned)K, 4u, 3u);
      tdm_load_2d(INb + (size_t)kb2 * J + j0,
                  LDS_OFF(&Bs[buf ^ 1][0][0]), 64u, 32u, (unsigned)J, 5u, 0u);
    }
#pragma unroll
    for (int kk = 0; kk < 32; kk += 4) {
      v2f a;
      a.x = Ws[buf][mrow][kk + koff];
      a.y = Ws[buf][mrow][kk + koff + 1];
#pragma unroll
      for (int jt = 0; jt < 4; ++jt) {
        v2f bf;
        bf.x = Bs[buf][kk + koff][jt * 16 + col];
        bf.y = Bs[buf][kk + koff + 1][jt * 16 + col];
        acc[jt] = wmma_f32(a, bf, acc[jt]);
      }
    }
    __syncthreads();
    buf ^= 1;
  }
#else
  for (int ib = 0; ib < nkb; ++ib) {
    int kb = ib << 5;
    for (int it = 0; it < 4; ++it) {  // W panel: 512 float4
      int idx = it * 128 + tid;
      int r = idx >> 3;
      int c4 = (idx & 7) * 4;
      float4 w4 = *reinterpret_cast<const float4*>(
          W + (size_t)(m0 + r) * K + kb + c4);
      Ws[0][r][c4 + 0] = w4.x; Ws[0][r][c4 + 1] = w4.y;
      Ws[0][r][c4 + 2] = w4.z; Ws[0][r][c4 + 3] = w4.w;
    }
    for (int it = 0; it < 4; ++it) {  // IN panel: 512 float4
      int idx = it * 128 + tid;
      int r = idx >> 4;
      int c4 = (idx & 15) * 4;
      float4 b4 = *reinterpret_cast<const float4*>(
          INb + (size_t)(kb + r) * J + j0 + c4);
      Bs[0][r][c4 + 0] = b4.x; Bs[0][r][c4 + 1] = b4.y;
      Bs[0][r][c4 + 2] = b4.z; Bs[0][r][c4 + 3] = b4.w;
    }
    if (kb + 32 < K) {
      __builtin_prefetch(W + (size_t)(m0 + (tid >> 1)) * K + kb + 32, 0, 0);
      __builtin_prefetch(INb + (size_t)(kb + 32 + (tid >> 2)) * J + j0, 0, 0);
    }
    __syncthreads();
#pragma unroll
    for (int kk = 0; kk < 32; kk += 4) {
      v2f a;
      a.x = Ws[0][mrow][kk + koff];
      a.y = Ws[0][mrow][kk + koff + 1];
#pragma unroll
      for (int jt = 0; jt < 4; ++jt) {
        v2f bf;
        bf.x = Bs[0][kk + koff][jt * 16 + col];
        bf.y = Bs[0][kk + koff + 1][jt * 16 + col];
        acc[jt] = wmma_f32(a, bf, acc[jt]);
      }
    }
    __syncthreads();
  }
#endif

  const int rbase = m0 + wave * 16 + ((lane >> 4) ? 8 : 0);
#pragma unroll
  for (int jt = 0; jt < 4; ++jt) {
#pragma unroll
    for (int i = 0; i < 8; ++i) {
      int m = rbase + i;
      int j = j0 + jt * 16 + col;
      float v = acc[jt][i];
      if (bias) v += bias[m];
      if (gamma) v = v * (gamma[m] * BN_SCALE) + beta[m];
      if (res) v += res[(size_t)b * M * J + (size_t)m * J + j];
      if (act) v = v > 0.0f ? v : 0.1f * v;
      OUT[(size_t)b * M * J + (size_t)m * J + j] = v;
    }
  }
}

// ---------------------------------------------------------------------------
// Positional encoding add: OUT = IN + pe(c, pos), pos = v (spatial) or t
// ---------------------------------------------------------------------------
__global__ void add_pe_kernel(const float* __restrict__ IN,
                              float* __restrict__ OUT, int spatial) {
  size_t idx = (size_t)blockIdx.x * 256 + threadIdx.x;
  if (idx >= (size_t)32 * 256 * 1600) return;
  int tv = (int)(idx % 1600);
  int c = (int)((idx / 1600) % 256);
  int t = tv / 25, v = tv - t * 25;
  float pos = spatial ? (float)v : (float)t;
  int i = c >> 1;
  float div = expf((float)(2 * i) * (-9.210340371976184f / 256.0f));
  float ang = pos * div;
  float pe = (c & 1) ? cosf(ang) : sinf(ang);
  OUT[idx] = IN[idx] + pe;
}

// ---------------------------------------------------------------------------
// muQ/muK means.  QKG layout: [n][which*192 + s*64 + c][t*25+v]
// ---------------------------------------------------------------------------
__global__ void mean_v_kernel(const float* __restrict__ QKG,
                              float* __restrict__ MU, int which) {
  int idx = blockIdx.x * 256 + threadIdx.x;  // N*S*64*64 = 393216
  if (idx >= 32 * 3 * 64 * 64) return;
  int t = idx & 63;
  int c = (idx >> 6) & 63;
  int hi = idx >> 12;
  int s = hi % 3, n = hi / 3;
  const float* p =
      QKG + ((size_t)n * 576 + which * 192 + s * 64 + c) * 1600 + t * 25;
  float a = 0.0f;
  for (int v = 0; v < 25; ++v) a += p[v];
  MU[idx] = a * (1.0f / 25.0f);
}

__global__ void mean_t_kernel(const float* __restrict__ QKG,
                              float* __restrict__ MU, int which) {
  int idx = blockIdx.x * 256 + threadIdx.x;  // N*S*64*25 = 153600
  if (idx >= 32 * 3 * 64 * 25) return;
  int v = idx % 25;
  int r = idx / 25;
  int c = r & 63;
  int hi = r >> 6;
  int s = hi % 3, n = hi / 3;
  const float* p =
      QKG + ((size_t)n * 576 + which * 192 + s * 64 + c) * 1600 + v;
  float a = 0.0f;
  for (int t = 0; t < 64; ++t) a += p[t * 25];
  MU[idx] = a * (1.0f / 64.0f);
}

// ---------------------------------------------------------------------------
// Spatial attention: per (n,s) compute att[32][32] (padded, zero for u/v>=25)
// Block = 64 thr (2 waves); K = IC*T = 4096, BK=32.
// ---------------------------------------------------------------------------
__global__ void __launch_bounds__(64) att_spatial_wmma(
    const float* __restrict__ QKG, const float* __restrict__ QKGP,
    const float* __restrict__ muQ, const float* __restrict__ muK,
    float* __restrict__ ATT) {
  const int s = blockIdx.x;
  const int n = blockIdx.y;
  const int tid = threadIdx.x;
  const int lane = tid & 31;
  const int wave = tid >> 5;
  __shared__ float Qs[32][33], Ks[32][33], Qps[32][33], Kps[32][33];
  __shared__ float un[32];

  const float sc = 1.0f / 4096.0f;
  const float* qptr = QKG + ((size_t)n * 576 + s * 64) * 1600;
  const float* kptr = QKG + ((size_t)n * 576 + 192 + s * 64) * 1600;
  const float* gptr = QKG + ((size_t)n * 576 + 384 + s * 64) * 1600;
  const float* qpptr = QKGP + ((size_t)n * 576 + s * 64) * 1600;
  const float* kpptr = QKGP + ((size_t)n * 576 + 192 + s * 64) * 1600;
  const float* mq = muQ + ((size_t)n * 3 + s) * 64 * 64;  // [c][t]
  const float* mk = muK + ((size_t)n * 3 + s) * 64 * 64;

  if (tid < 32) {
    float a = 0.0f;
    if (tid < 25) {
      for (int Kx = 0; Kx < 4096; ++Kx) {
        int c = Kx >> 6, t = Kx & 63;
        a += mq[c * 64 + t] * gptr[(size_t)c * 1600 + t * 25 + tid];
      }
    }
    un[tid] = a * sc;
  }

  v8f accP[2] = {};
  v8f accO[2] = {};
  const int col = lane & 15;
  const int koff = (lane >> 4) * 2;
  const int u0 = wave * 16;

  for (int kb = 0; kb < 4096; kb += 32) {
    for (int it = 0; it < 16; ++it) {
      int idx = it * 64 + tid;  // 0..1023
      int kl = idx >> 5;
      int u = idx & 31;
      int Kx = kb + kl;
      int c = Kx >> 6, t = Kx & 63;
      float qv = 0.0f, kv = 0.0f, qpv = 0.0f, kpv = 0.0f;
      if (u < 25) {
        size_t base = (size_t)c * 1600 + t * 25 + u;
        float mqv = mq[c * 64 + t], mkv = mk[c * 64 + t];
        qv = qptr[base] - mqv;
        kv = kptr[base] - mkv;
        qpv = qpptr[base];
        kpv = kpptr[base];
      }
      Qs[kl][u] = qv;
      Ks[kl][u] = kv;
      Qps[kl][u] = qpv;
      Kps[kl][u] = kpv;
    }
    __syncthreads();
#pragma unroll
    for (int kk = 0; kk < 32; kk += 4) {
      v2f aq, ap;
      aq.x = Qs[kk + koff][u0 + col];
      aq.y = Qs[kk + koff + 1][u0 + col];
      ap.x = Qps[kk + koff][u0 + col];
      ap.y = Qps[kk + koff + 1][u0 + col];
#pragma unroll
      for (int vt = 0; vt < 2; ++vt) {
        v2f bk, bp;
        bk.x = Ks[kk + koff][vt * 16 + col];
        bk.y = Ks[kk + koff + 1][vt * 16 + col];
        accP[vt] = wmma_f32(aq, bk, accP[vt]);
        bp.x = Kps[kk + koff][vt * 16 + col];
        bp.y = Kps[kk + koff + 1][vt * 16 + col];
        accO[vt] = wmma_f32(ap, bp, accO[vt]);
      }
    }
    __syncthreads();
  }

  float* att = ATT + ((size_t)n * 3 + s) * 1024;
#pragma unroll
  for (int vt = 0; vt < 2; ++vt)
#pragma unroll
    for (int i = 0; i < 8; ++i) {
      int u = u0 + i + ((lane >> 4) ? 8 : 0);
      int v = vt * 16 + col;
      float pw = accP[vt][i] * sc;
      float po = accO[vt][i] * sc;
      float val = tanhf(pw + po) + tanhf(un[v] + po);
      if (u >= 25 || v >= 25) val = 0.0f;  // zero K/N padding
      att[u * 32 + v] = val;
    }
}

// ---------------------------------------------------------------------------
// Temporal attention: per (n,s) compute att[64][64]; K = IC*V = 1600, BK=32.
// Block = 128 thr (4 waves).
// ---------------------------------------------------------------------------
__global__ void __launch_bounds__(128) att_temporal_wmma(
    const float* __restrict__ QKG, const float* __restrict__ QKGP,
    const float* __restrict__ muQ, const float* __restrict__ muK,
    float* __restrict__ ATT) {
  const int s = blockIdx.x;
  const int n = blockIdx.y;
  const int tid = threadIdx.x;
  const int lane = tid & 31;
  const int wave = tid >> 5;
  __shared__ float Qs[32][65], Ks[32][65], Qps[32][65], Kps[32][65];
  __shared__ float un[64];

  const float sct = 1.0f / 1600.0f;
  const float* qptr = QKG + ((size_t)n * 576 + s * 64) * 1600;
  const float* kptr = QKG + ((size_t)n * 576 + 192 + s * 64) * 1600;
  const float* gptr = QKG + ((size_t)n * 576 + 384 + s * 64) * 1600;
  const float* qpptr = QKGP + ((size_t)n * 576 + s * 64) * 1600;
  const float* kpptr = QKGP + ((size_t)n * 576 + 192 + s * 64) * 1600;
  const float* mq = muQ + ((size_t)n * 3 + s) * 64 * 25;  // [c][v]
  const float* mk = muK + ((size_t)n * 3 + s) * 64 * 25;

  if (tid < 64) {
    float a = 0.0f;
    for (int Kx = 0; Kx < 1600; ++Kx) {
      int c = Kx / 25, v = Kx - c * 25;
      a += mq[c * 25 + v] * gptr[(size_t)c * 1600 + tid * 25 + v];
    }
    un[tid] = a * sct;
  }

  v8f accP[4] = {};
  v8f accO[4] = {};
  const int col = lane & 15;
  const int koff = (lane >> 4) * 2;
  const int t0 = wave * 16;

  for (int kb = 0; kb < 1600; kb += 32) {
    for (int it = 0; it < 16; ++it) {
      int idx = it * 128 + tid;  // 0..2047
      int kl = idx >> 6;
      int t = idx & 63;
      int Kx = kb + kl;
      int c = Kx / 25, v = Kx - c * 25;
      size_t base = (size_t)c * 1600 + t * 25 + v;
      Qs[kl][t] = qptr[base] - mq[c * 25 + v];
      Ks[kl][t] = kptr[base] - mk[c * 25 + v];
      Qps[kl][t] = qpptr[base];
      Kps[kl][t] = kpptr[base];
    }
    __syncthreads();
#pragma unroll
    for (int kk = 0; kk < 32; kk += 4) {
      v2f aq, ap;
      aq.x = Qs[kk + koff][t0 + col];
      aq.y = Qs[kk + koff + 1][t0 + col];
      ap.x = Qps[kk + koff][t0 + col];
      ap.y = Qps[kk + koff + 1][t0 + col];
#pragma unroll
      for (int qt = 0; qt < 4; ++qt) {
        v2f bk, bp;
        bk.x = Ks[kk + koff][qt * 16 + col];
        bk.y = Ks[kk + koff + 1][qt * 16 + col];
        accP[qt] = wmma_f32(aq, bk, accP[qt]);
        bp.x = Kps[kk + koff][qt * 16 + col];
        bp.y = Kps[kk + koff + 1][qt * 16 + col];
        accO[qt] = wmma_f32(ap, bp, accO[qt]);
      }
    }
    __syncthreads();
  }

  float* att = ATT + ((size_t)n * 3 + s) * 4096;
#pragma unroll
  for (int qt = 0; qt < 4; ++qt)
#pragma unroll
    for (int i = 0; i < 8; ++i) {
      int t = t0 + i + ((lane >> 4) ? 8 : 0);
      int q = qt * 16 + col;
      float pw = accP[qt][i] * sct;
      float po = accO[qt][i] * sct;
      att[t * 64 + q] = tanhf(pw + po) + tanhf(un[q] + po);
    }
}

// ---------------------------------------------------------------------------
// Apply spatial attention: YPRE[n][s*256+c][t*25+v] = sum_u x[n,c,t,u]*att[u,v]
// Rows m=(c,t): M=16384; K=25 (padded 32); N=25 (padded 32).
// ---------------------------------------------------------------------------
__global__ void __launch_bounds__(128) apply_att_spatial(
    const float* __restrict__ X, const float* __restrict__ ATT,
    float* __restrict__ YPRE) {
  const int blk = blockIdx.x;  // 0..255
  const int s = blockIdx.y;
  const int n = blockIdx.z;
  const int tid = threadIdx.x;
  const int lane = tid & 31;
  const int wave = tid >> 5;
  __shared__ float Bs[32][33];

  const float* att = ATT + ((size_t)n * 3 + s) * 1024;
#if HAVE_TDM
  if (wave == 0) {
    tdm_load_2d(att, LDS_OFF(&Bs[0][0]), 32u, 32u, 32u, 4u, 0u);
    __builtin_amdgcn_s_wait_tensorcnt(0);
  }
  __syncthreads();
#else
  for (int it = 0; it < 8; ++it) {
    int idx = it * 128 + tid;
    Bs[idx >> 5][idx & 31] = att[idx];
  }
  __syncthreads();
#endif

  const int col = lane & 15;
  const int koff = (lane >> 4) * 2;
  const int m = blk * 64 + wave * 16 + col;
  const int c = m >> 6, t = m & 63;
  const float* xr = X + (size_t)n * 256 * 1600 + (size_t)c * 1600 + t * 25;

  v8f acc[2] = {};
#pragma unroll
  for (int kk = 0; kk < 32; kk += 4) {
    int kbase = kk + koff;
    v2f a;
    a.x = (kbase < 25) ? xr[kbase] : 0.0f;
    a.y = (kbase + 1 < 25) ? xr[kbase + 1] : 0.0f;
#pragma unroll
    for (int vt = 0; vt < 2; ++vt) {
      v2f b;
      b.x = Bs[kbase][vt * 16 + col];
      b.y = Bs[kbase + 1][vt * 16 + col];
      acc[vt] = wmma_f32(a, b, acc[vt]);
    }
  }

  const int rb = blk * 64 + wave * 16 + ((lane >> 4) ? 8 : 0);
#pragma unroll
  for (int vt = 0; vt < 2; ++vt)
#pragma unroll
    for (int i = 0; i < 8; ++i) {
      int mm = rb + i;
      int cc = mm >> 6, tt = mm & 63;
      int v = vt * 16 + col;
      if (v < 25)
        YPRE[(size_t)n * 768 * 1600 + (size_t)(s * 256 + cc) * 1600 + tt * 25 +
             v] = acc[vt][i];
    }
}

// ---------------------------------------------------------------------------
// Apply temporal attention: ZPRE[n][s*256+c][q*25+v] = sum_t y[n,c,t,v]*att[t,q]
// Rows m=(c,v): M=6400; K=64; N=64.
// ---------------------------------------------------------------------------
__global__ void __launch_bounds__(128) apply_att_temporal(
    const float* __restrict__ Y, const float* __restrict__ ATT,
    float* __restrict__ ZPRE) {
  const int blk = blockIdx.x;  // 0..99
  const int s = blockIdx.y;
  const int n = blockIdx.z;
  const int tid = threadIdx.x;
  const int lane = tid & 31;
  const int wave = tid >> 5;
  __shared__ float Bs[64][65];

  const float* att = ATT + ((size_t)n * 3 + s) * 4096;
#if HAVE_TDM
  if (wave == 0) {
    tdm_load_2d(att, LDS_OFF(&Bs[0][0]), 64u, 64u, 64u, 5u, 0u);
    __builtin_amdgcn_s_wait_tensorcnt(0);
  }
  __syncthreads();
#else
  for (int it = 0; it < 32; ++it) {
    int idx = it * 128 + tid;
    Bs[idx >> 6][idx & 63] = att[idx];
  }
  __syncthreads();
#endif

  const int col = lane & 15;
  const int koff = (lane >> 4) * 2;
  const int m = blk * 64 + wave * 16 + col;
  const int c = m / 25, v = m - c * 25;
  const float* yr = Y + (size_t)n * 256 * 1600 + (size_t)c * 1600 + v;

  v8f acc[4] = {};
#pragma unroll
  for (int kk = 0; kk < 64; kk += 4) {
    int kb = kk + koff;
    v2f a;
    a.x = yr[kb * 25];
    a.y = yr[(kb + 1) * 25];
#pragma unroll
    for (int qt = 0; qt < 4; ++qt) {
      v2f b;
      b.x = Bs[kb][qt * 16 + col];
      b.y = Bs[kb + 1][qt * 16 + col];
      acc[qt] = wmma_f32(a, b, acc[qt]);
    }
  }

  const int rb = blk * 64 + wave * 16 + ((lane >> 4) ? 8 : 0);
#pragma unroll
  for (int qt = 0; qt < 4; ++qt)
#pragma unroll
    for (int i = 0; i < 8; ++i) {
      int mm = rb + i;
      int cc = mm / 25, vv = mm - cc * 25;
      int q = qt * 16 + col;
      ZPRE[(size_t)n * 768 * 1600 + (size_t)(s * 256 + cc) * 1600 + q * 25 +
           vv] = acc[qt][i];
    }
}

// ---------------------------------------------------------------------------
extern "C" void kernel_launch(void* const* d_in, const int* in_sizes, int n_in,
                              void* d_out, int out_size, void* d_ws,
                              size_t ws_size, hipStream_t stream) {
  const float* x = (const float*)d_in[0];
  const float* Win_s = (const float*)d_in[1];
  const float* bin_s = (const float*)d_in[2];
  const float* Wout_s = (const float*)d_in[3];
  const float* bout_s = (const float*)d_in[4];
  const float* gout_s = (const float*)d_in[5];
  const float* bnout_s = (const float*)d_in[6];
  const float* Wff_s = (const float*)d_in[7];
  const float* bff_s = (const float*)d_in[8];
  const float* gff_s = (const float*)d_in[9];
  const float* bnff_s = (const float*)d_in[10];
  const float* Win_t = (const float*)d_in[11];
  const float* bin_t = (const float*)d_in[12];
  const float* Wout_t = (const float*)d_in[13];
  const float* bout_t = (const float*)d_in[14];
  const float* gout_t = (const float*)d_in[15];
  const float* bnout_t = (const float*)d_in[16];
  const float* Wff_t = (const float*)d_in[17];
  const float* bff_t = (const float*)d_in[18];
  const float* gff_t = (const float*)d_in[19];
  const float* bnff_t = (const float*)d_in[20];
  float* out = (float*)d_out;

  // Workspace layout (floats). YPRE aliases QKG+QKGP region (safe: stream-
  // serialized, QKG/QKGP dead once attention matrices are computed).
  float* ws = (float*)d_ws;
  size_t o = 0;
  float* bufPE = ws + o;   o += 13107200;  // [N,256,1600]
  float* bufQKG = ws + o;  o += 29491200;  // [N,576,1600]
  float* bufQKGP = ws + o; o += 29491200;  // [N,576,1600]
  float* bufMUQ = ws + o;  o += 393216;
  float* bufMUK = ws + o;  o += 393216;
  float* bufATT = ws + o;  o += 393216;    // max(N*S*32*32, N*S*64*64)
  float* bufY1 = ws + o;   o += 13107200;
  float* bufY = ws + o;    o += 13107200;
  float* bufYPRE = bufQKG;                 // [N,768,1600] alias

  const int NEL = 13107200;

  // ---------------- spatial ----------------
  add_pe_kernel<<<(NEL + 255) / 256, 256, 0, stream>>>(x, bufPE, 1);
  conv1x1_wmma<<<dim3(25, 9, 32), 128, 0, stream>>>(
      Win_s, x, bufQKG, bin_s, nullptr, nullptr, nullptr, 576, 256, 1600, 0);
  conv1x1_wmma<<<dim3(25, 9, 32), 128, 0, stream>>>(
      Win_s, bufPE, bufQKGP, bin_s, nullptr, nullptr, nullptr, 576, 256, 1600,
      0);
  mean_v_kernel<<<(393216 + 255) / 256, 256, 0, stream>>>(bufQKG, bufMUQ, 0);
  mean_v_kernel<<<(393216 + 255) / 256, 256, 0, stream>>>(bufQKG, bufMUK, 1);
  att_spatial_wmma<<<dim3(3, 32), 64, 0, stream>>>(bufQKG, bufQKGP, bufMUQ,
                                                   bufMUK, bufATT);
  apply_att_spatial<<<dim3(256, 3, 32), 128, 0, stream>>>(x, bufATT, bufYPRE);
  conv1x1_wmma<<<dim3(25, 4, 32), 128, 0, stream>>>(
      Wout_s, bufYPRE, bufY1, bout_s, gout_s, bnout_s, x, 256, 768, 1600, 1);
  conv1x1_wmma<<<dim3(25, 4, 32), 128, 0, stream>>>(
      Wff_s, bufY1, bufY, bff_s, gff_s, bnff_s, x, 256, 256, 1600, 1);

  // ---------------- temporal ----------------
  add_pe_kernel<<<(NEL + 255) / 256, 256, 0, stream>>>(bufY, bufPE, 0);
  conv1x1_wmma<<<dim3(25, 9, 32), 128, 0, stream>>>(
      Win_t, bufY, bufQKG, bin_t, nullptr, nullptr, nullptr, 576, 256, 1600, 0);
  conv1x1_wmma<<<dim3(25, 9, 32), 128, 0, stream>>>(
      Win_t, bufPE, bufQKGP, bin_t, nullptr, nullptr, nullptr, 576, 256, 1600,
      0);
  mean_t_kernel<<<(153600 + 255) / 256, 256, 0, stream>>>(bufQKG, bufMUQ, 0);
  mean_t_kernel<<<(153600 + 255) / 256, 256, 0, stream>>>(bufQKG, bufMUK, 1);
  att_temporal_wmma<<<dim3(3, 32), 128, 0, stream>>>(bufQKG, bufQKGP, bufMUQ,
                                                     bufMUK, bufATT);
  apply_att_temporal<<<dim3(100, 3, 32), 128, 0, stream>>>(bufY, bufATT,
                                                           bufYPRE);
  conv1x1_wmma<<<dim3(25, 4, 32), 128, 0, stream>>>(
      Wout_t, bufYPRE, bufY1, bout_t, gout_t, bnout_t, bufY, 256, 768, 1600, 1);
  conv1x1_wmma<<<dim3(25, 4, 32), 128, 0, stream>>>(
      Wff_t, bufY1, out, bff_t, gff_t, bnff_t, bufY, 256, 256, 1600, 1);
}